// ColBERT_85461259256261
// MI455X (gfx1250) — compile-verified
//
#include <hip/hip_runtime.h>
#include <cmath>

// ---------------------------------------------------------------------------
// ColBERT MaxSim on gfx1250 (MI455X), fp32 WMMA path.
//   q = l2norm(q_hidden @ W^T)   [32,32,128]
//   d = l2norm(d_hidden @ W^T)   [256,512,128]
//   out[b] = sum_q max_k ( qd[b] . d[b] ) with -inf doc-token masking
// ---------------------------------------------------------------------------

typedef __attribute__((ext_vector_type(2))) float v2f;
typedef __attribute__((ext_vector_type(8))) float v8f;

#define HDIM 768
#define DIM  128
#define BQ   32
#define LQ   32
#define BD   256
#define LD   512
// ppq fixed at 8 by the reference setup (d_in[4] carries it but shapes are static)

// ---------------------------------------------------------------------------
// Kernel 1: Y[t, :] = l2norm( X[t, :] @ W^T ),  X:[T,768], W:[128,768], Y:[T,128]
// Block = 256 threads = 8 waves. Block handles 16 token rows; wave w handles
// output columns [16w, 16w+16). K consumed 4 at a time via v_wmma_f32_16x16x4.
//
// fp32 WMMA fragment layout (ISA 7.12.2, 32-bit A 16x4):
//   lane l in [0,16):  v0 = A[M=l][K=0], v1 = A[l][1]
//   lane l in [16,32): v0 = A[l-16][2],  v1 = A[l-16][3]
// B (4x16) mirrors this with N striped across lanes. Both reduce to one
// float2 load per lane at (row, k0 + 2*half).
// ---------------------------------------------------------------------------
__global__ __launch_bounds__(256) void proj_l2norm_kernel(
    const float* __restrict__ X, const float* __restrict__ W,
    float* __restrict__ Y)
{
    __shared__ float rowSumSq[16];

    const int wave = threadIdx.x >> 5;
    const int lane = threadIdx.x & 31;
    const int half = lane >> 4;      // 0: K={0,1}, 1: K={2,3}
    const int l    = lane & 15;      // M (for A) / N (for B) within tile
    const int row0 = blockIdx.x * 16;
    const int n0   = wave * 16;

    if (threadIdx.x < 16) rowSumSq[threadIdx.x] = 0.0f;
    __syncthreads();

    const float* arow = X + (size_t)(row0 + l) * HDIM + 2 * half;
    const float* brow = W + (size_t)(n0   + l) * HDIM + 2 * half;

    v8f acc = {};
#pragma unroll 8
    for (int k0 = 0; k0 < HDIM; k0 += 4) {
        v2f a = *reinterpret_cast<const v2f*>(arow + k0);
        v2f b = *reinterpret_cast<const v2f*>(brow + k0);
        acc = __builtin_amdgcn_wmma_f32_16x16x4_f32(
            /*neg_a=*/false, a, /*neg_b=*/false, b,
            /*c_mod=*/(short)0, acc, /*reuse_a=*/false, /*reuse_b=*/false);
    }
    // acc[r] @ lane = Y_tile[M = r + 8*half][N = l]

    // Partial sum-of-squares per row: reduce over the 16 lanes of this half.
    float p[8];
#pragma unroll
    for (int r = 0; r < 8; ++r) {
        float v = acc[r] * acc[r];
        v += __shfl_xor(v, 1, 32);
        v += __shfl_xor(v, 2, 32);
        v += __shfl_xor(v, 4, 32);
        v += __shfl_xor(v, 8, 32);
        p[r] = v;
    }
    if (l == 0) {  // one lane per half accumulates its 8 rows across waves
#pragma unroll
        for (int r = 0; r < 8; ++r)
            atomicAdd(&rowSumSq[r + 8 * half], p[r]);  // ds_add_f32
    }
    __syncthreads();

#pragma unroll
    for (int r = 0; r < 8; ++r) {
        const int m   = r + 8 * half;
        const float nrm   = sqrtf(rowSumSq[m]);
        const float scale = 1.0f / fmaxf(nrm, 1e-12f);   // F.normalize eps
        Y[(size_t)(row0 + m) * DIM + n0 + l] = acc[r] * scale;
    }
}

// ---------------------------------------------------------------------------
// Kernel 2: per-doc MaxSim. One block per doc b (256 blocks), 8 waves.
// sim[q,k] = Qp[b/8][q,:] . Dp[b][k,:]  (M=32 queries, N=512 tokens, K=128)
// Wave w sweeps n-tiles {w, w+8, w+16, w+24}; two accumulators cover query
// rows 0-15 and 16-31 while sharing each B fragment.
// ---------------------------------------------------------------------------
__global__ __launch_bounds__(256) void maxsim_kernel(
    const float* __restrict__ Qp, const float* __restrict__ Dp,
    const unsigned char* __restrict__ d_mask, float* __restrict__ out)
{
    __shared__ float wmax[32][8];   // [query row][wave]

    const int b    = blockIdx.x;
    const int g    = b >> 3;        // ppq = 8
    const int wave = threadIdx.x >> 5;
    const int lane = threadIdx.x & 31;
    const int half = lane >> 4;
    const int l    = lane & 15;

    const float* qbase = Qp + (size_t)g * LQ * DIM;
    const float* dbase = Dp + (size_t)b * LD * DIM;
    const unsigned char* mrow = d_mask + (size_t)b * LD;

    const float NEG_INF = -__builtin_inff();
    float vmax0[8], vmax1[8];
#pragma unroll
    for (int r = 0; r < 8; ++r) { vmax0[r] = NEG_INF; vmax1[r] = NEG_INF; }

    const float* a0row = qbase + (size_t)(l)      * DIM + 2 * half;  // queries 0-15
    const float* a1row = qbase + (size_t)(16 + l) * DIM + 2 * half;  // queries 16-31

    for (int nt = wave; nt < LD / 16; nt += 8) {
        const int n0 = nt * 16;
        const float* brow = dbase + (size_t)(n0 + l) * DIM + 2 * half;

        v8f acc0 = {}, acc1 = {};
#pragma unroll
        for (int k0 = 0; k0 < DIM; k0 += 4) {
            v2f bf  = *reinterpret_cast<const v2f*>(brow  + k0);
            v2f af0 = *reinterpret_cast<const v2f*>(a0row + k0);
            v2f af1 = *reinterpret_cast<const v2f*>(a1row + k0);
            acc0 = __builtin_amdgcn_wmma_f32_16x16x4_f32(
                false, af0, false, bf, (short)0, acc0, false, false);
            acc1 = __builtin_amdgcn_wmma_f32_16x16x4_f32(
                false, af1, false, bf, (short)0, acc1, false, false);
        }

        // This lane owns doc-token column n0+l; mask -> -inf (reference semantics).
        const bool valid = (mrow[n0 + l] != 0);
#pragma unroll
        for (int r = 0; r < 8; ++r) {
            vmax0[r] = fmaxf(vmax0[r], valid ? acc0[r] : NEG_INF);
            vmax1[r] = fmaxf(vmax1[r], valid ? acc1[r] : NEG_INF);
        }
    }

    // Max across the 16 token columns held by this half-wave.
#pragma unroll
    for (int r = 0; r < 8; ++r) {
        float v0 = vmax0[r], v1 = vmax1[r];
#pragma unroll
        for (int m = 1; m <= 8; m <<= 1) {
            v0 = fmaxf(v0, __shfl_xor(v0, m, 32));
            v1 = fmaxf(v1, __shfl_xor(v1, m, 32));
        }
        vmax0[r] = v0; vmax1[r] = v1;
    }
    if (l == 0) {
#pragma unroll
        for (int r = 0; r < 8; ++r) {
            wmax[r + 8 * half][wave]      = vmax0[r];  // queries 0-15
            wmax[16 + r + 8 * half][wave] = vmax1[r];  // queries 16-31
        }
    }
    __syncthreads();

    if (threadIdx.x < 32) {
        float m = NEG_INF;
#pragma unroll
        for (int w = 0; w < 8; ++w) m = fmaxf(m, wmax[threadIdx.x][w]);
        // Sum of per-query maxima across all 32 lanes (wave32 full reduce).
        m += __shfl_xor(m, 1, 32);
        m += __shfl_xor(m, 2, 32);
        m += __shfl_xor(m, 4, 32);
        m += __shfl_xor(m, 8, 32);
        m += __shfl_xor(m, 16, 32);
        if (threadIdx.x == 0) out[b] = m;
    }
}

// ---------------------------------------------------------------------------
extern "C" void kernel_launch(void* const* d_in, const int* in_sizes, int n_in,
                              void* d_out, int out_size, void* d_ws, size_t ws_size,
                              hipStream_t stream)
{
    const float*         q_hidden = (const float*)d_in[0];          // [32,32,768]
    const float*         d_hidden = (const float*)d_in[1];          // [256,512,768]
    const float*         W        = (const float*)d_in[2];          // [128,768]
    const unsigned char* d_mask   = (const unsigned char*)d_in[3];  // [256,512] bool
    // d_in[4] = ppq (8; shapes are static)
    float* out = (float*)d_out;                                     // [256]

    float* q_proj = (float*)d_ws;                        // 1024*128 f32 (512 KB)
    float* d_proj = q_proj + (size_t)BQ * LQ * DIM;      // 131072*128 f32 (64 MB)

    proj_l2norm_kernel<<<(BQ * LQ) / 16, 256, 0, stream>>>(q_hidden, W, q_proj);
    proj_l2norm_kernel<<<(BD * LD) / 16, 256, 0, stream>>>(d_hidden, W, d_proj);
    maxsim_kernel<<<BD, 256, 0, stream>>>(q_proj, d_proj, d_mask, out);
}